// ContinuousTimeHopfieldLayer_6493990551907
// MI455X (gfx1250) — compile-verified
//
#include <hip/hip_runtime.h>
#include <hip/hip_bf16.h>

typedef __attribute__((ext_vector_type(2))) float v2f;
typedef __attribute__((ext_vector_type(8))) float v8f;

#define UNITS 256
#define BATCH 32768
#define MTILE 16
#define LDA   260   // padded LDS row stride in floats (260 mod 64 == 4 -> conflict-free A reads)

#define TAUINV    1.0f
#define DT0       0.1f
#define RK_TOL    1e-3f
#define MIN_DT    0.01f
#define MAX_DT    1.0f
#define MAX_TIME  1000.0f
#define RELAX_TOL 0.01f
#define MAX_STEPS 40

// control block word indices
#define C_T     0
#define C_DT    1
#define C_CUR   2
#define C_DONE  3
#define C_ERR   4
#define C_RELAX 5

// ---------------------------------------------------------------- init / prep

__global__ void init_ctrl(unsigned* __restrict__ ctrl) {
  if (threadIdx.x == 0) {
    ctrl[C_T]     = __float_as_uint(0.0f);
    ctrl[C_DT]    = __float_as_uint(DT0);
    ctrl[C_CUR]   = 0u;
    ctrl[C_DONE]  = 0u;
    ctrl[C_ERR]   = 0u;
    ctrl[C_RELAX] = 0u;
  }
}

// Pack Ws = sym-zero-diag(W) directly into WMMA B-fragment order:
// P2[((k0/4)*2 + half)*UNITS + n] = { Ws[k0+2*half][n], Ws[k0+2*half+1][n] }
// so each lane's B fragment is one contiguous 8-byte load.
__global__ void prep_pack(const float* __restrict__ W, float2* __restrict__ P) {
  const int g = blockIdx.x * blockDim.x + threadIdx.x;   // 0 .. UNITS*UNITS/2 - 1
  const int n = g & (UNITS - 1);
  const int pairIdx = g >> 8;                            // 0 .. 127
  const int k = (pairIdx >> 1) * 4 + (pairIdx & 1) * 2;  // K row of .x element
  float a0 = (k     == n) ? 0.0f : 0.5f * (W[(k    ) * UNITS + n] + W[n * UNITS + k    ]);
  float a1 = (k + 1 == n) ? 0.0f : 0.5f * (W[(k + 1) * UNITS + n] + W[n * UNITS + k + 1]);
  P[(size_t)pairIdx * UNITS + n] = make_float2(a0, a1);
}

__global__ void copy_in(const float4* __restrict__ src, float4* __restrict__ dst, long n4) {
  long i = (long)blockIdx.x * blockDim.x + threadIdx.x;
  if (i < n4) dst[i] = src[i];
}

// ---------------------------------------------------------------- RKF stage

// Stage J (0-based): builds xt = x + dt * sum_i A[J][i] * k_i, computes
// k_J = (tanh(xt @ Ws + b) - xt)/tau via f32 WMMA, redistributes k_J into
// per-thread registers. `tile` is the shared staging buffer (xt, then k_J).
template <int J>
__device__ __forceinline__ void run_stage(
    const v2f* __restrict__ Pv, const float* __restrict__ bvec,
    float (&tile)[MTILE][LDA],
    const float (&xr)[16], float (&kr)[6][16],
    float dt, int mrow, int c0, int wid, int half, int lm)
{
  // ---- build xt into LDS from per-thread registers (elementwise ownership)
  #pragma unroll
  for (int e = 0; e < 16; ++e) {
    float s = 0.0f;
    if constexpr (J == 1) {
      s = 0.25f * kr[0][e];
    } else if constexpr (J == 2) {
      s = (3.0f/32.0f) * kr[0][e] + (9.0f/32.0f) * kr[1][e];
    } else if constexpr (J == 3) {
      s = (1932.0f/2197.0f) * kr[0][e] + (-7200.0f/2197.0f) * kr[1][e]
        + (7296.0f/2197.0f) * kr[2][e];
    } else if constexpr (J == 4) {
      s = (439.0f/216.0f) * kr[0][e] + (-8.0f) * kr[1][e]
        + (3680.0f/513.0f) * kr[2][e] + (-845.0f/4104.0f) * kr[3][e];
    } else if constexpr (J == 5) {
      s = (-8.0f/27.0f) * kr[0][e] + (2.0f) * kr[1][e]
        + (-3544.0f/2565.0f) * kr[2][e] + (1859.0f/4104.0f) * kr[3][e]
        + (-11.0f/40.0f) * kr[4][e];
    }
    tile[mrow][c0 + e] = xr[e] + dt * s;
  }
  __syncthreads();

  // ---- GEMM: each wave owns two 16x16 column tiles of the 16x256 output
  const int n0a = wid * 32;
  const int n0b = wid * 32 + 16;
  const int baseA = n0a + lm;
  const int baseB = n0b + lm;
  v8f acc0 = {};
  v8f acc1 = {};
  #pragma unroll 8
  for (int k0 = 0; k0 < UNITS; k0 += 4) {
    const int ka = k0 + 2 * half;       // A: lanes 0-15 K={k0,k0+1}; lanes 16-31 K={k0+2,k0+3}
    v2f a;
    a.x = tile[lm][ka];
    a.y = tile[lm][ka + 1];
    const int pbase = ((k0 >> 2) * 2 + half) * UNITS;   // packed B fragment row
    v2f b0 = Pv[pbase + baseA];
    v2f b1 = Pv[pbase + baseB];
    acc0 = __builtin_amdgcn_wmma_f32_16x16x4_f32(false, a, false, b0,
                                                 (short)0, acc0, false, false);
    acc1 = __builtin_amdgcn_wmma_f32_16x16x4_f32(false, a, false, b1,
                                                 (short)0, acc1, false, false);
  }

  // ---- epilogue: k = (tanh(acc + b) - xt)/tau, in C layout
  float kv[16];
  #pragma unroll
  for (int r = 0; r < 8; ++r) {
    const int m = r + 8 * half;   // C VGPR r: lanes 0-15 -> row r, lanes 16-31 -> row r+8
    {
      const int n = n0a + lm;
      float v = tanhf(acc0[r] + bvec[n]);
      kv[r] = (v - tile[m][n]) * TAUINV;
    }
    {
      const int n = n0b + lm;
      float v = tanhf(acc1[r] + bvec[n]);
      kv[8 + r] = (v - tile[m][n]) * TAUINV;
    }
  }
  __syncthreads();                     // all xt reads done; tile becomes k buffer

  #pragma unroll
  for (int r = 0; r < 8; ++r) {
    tile[r + 8 * half][n0a + lm] = kv[r];
    tile[r + 8 * half][n0b + lm] = kv[8 + r];
  }
  __syncthreads();

  #pragma unroll
  for (int e = 0; e < 16; ++e) kr[J][e] = tile[mrow][c0 + e];
  __syncthreads();                     // reads done before next stage overwrites tile
}

// ---------------------------------------------------------------- fused RKF step

__launch_bounds__(256)
__global__ void hopfield_rkf_step(const float2* __restrict__ P,
                                  const float* __restrict__ bvec,
                                  float* __restrict__ buf0,
                                  float* __restrict__ buf1,
                                  unsigned* __restrict__ ctrl)
{
  if (ctrl[C_DONE]) return;            // uniform early-exit once converged
  const unsigned cur = ctrl[C_CUR];
  const float dt = __uint_as_float(ctrl[C_DT]);
  const float* __restrict__ xin  = cur ? buf1 : buf0;
  float* __restrict__ xout       = cur ? buf0 : buf1;
  const v2f* __restrict__ Pv = (const v2f*)P;

  __shared__ float tile[MTILE][LDA];
  __shared__ unsigned red_s;

  const int tid  = threadIdx.x;
  const int wid  = tid >> 5;
  const int lane = tid & 31;
  const int half = lane >> 4;
  const int lm   = lane & 15;
  const int mrow = tid >> 4;           // owned row within tile (0..15)
  const int c0   = (tid & 15) * 16;    // owned col base (16 consecutive cols)
  const long row0 = (long)blockIdx.x * MTILE;

  float xr[16];
  float kr[6][16];
  #pragma unroll
  for (int e = 0; e < 16; ++e)
    xr[e] = xin[(row0 + mrow) * UNITS + c0 + e];

  // ---- k1 = pvf(x); also the relaxation norm source (cond check)
  run_stage<0>(Pv, bvec, tile, xr, kr, dt, mrow, c0, wid, half, lm);

  {
    float rl = 0.0f;
    #pragma unroll
    for (int e = 0; e < 16; ++e) rl = fmaxf(rl, fabsf(kr[0][e]));
    if (tid == 0) red_s = 0u;
    __syncthreads();
    atomicMax(&red_s, __float_as_uint(rl));
    __syncthreads();
    if (tid == 0) atomicMax(&ctrl[C_RELAX], red_s);
    __syncthreads();
  }

  run_stage<1>(Pv, bvec, tile, xr, kr, dt, mrow, c0, wid, half, lm);
  run_stage<2>(Pv, bvec, tile, xr, kr, dt, mrow, c0, wid, half, lm);
  run_stage<3>(Pv, bvec, tile, xr, kr, dt, mrow, c0, wid, half, lm);
  run_stage<4>(Pv, bvec, tile, xr, kr, dt, mrow, c0, wid, half, lm);
  run_stage<5>(Pv, bvec, tile, xr, kr, dt, mrow, c0, wid, half, lm);

  // ---- 4th/5th order combine, err = max|x5-x4|, write speculative x5
  float errl = 0.0f;
  #pragma unroll
  for (int e = 0; e < 16; ++e) {
    const float k1 = kr[0][e], k3 = kr[2][e], k4 = kr[3][e],
                k5 = kr[4][e], k6 = kr[5][e];
    const float x4 = xr[e] + dt * ((25.0f/216.0f) * k1 + (1408.0f/2565.0f) * k3
                                 + (2197.0f/4104.0f) * k4 - 0.2f * k5);
    const float x5 = xr[e] + dt * ((16.0f/135.0f) * k1 + (6656.0f/12825.0f) * k3
                                 + (28561.0f/56430.0f) * k4 - (9.0f/50.0f) * k5
                                 + (2.0f/55.0f) * k6);
    errl = fmaxf(errl, fabsf(x5 - x4));
    xout[(row0 + mrow) * UNITS + c0 + e] = x5;
  }
  if (tid == 0) red_s = 0u;
  __syncthreads();
  atomicMax(&red_s, __float_as_uint(errl));
  __syncthreads();
  if (tid == 0) atomicMax(&ctrl[C_ERR], red_s);
}

// ---------------------------------------------------------------- control step

__global__ void rkf_control(unsigned* __restrict__ ctrl) {
  if (threadIdx.x != 0 || blockIdx.x != 0) return;
  if (ctrl[C_DONE]) return;

  const float t     = __uint_as_float(ctrl[C_T]);
  const float relax = __uint_as_float(ctrl[C_RELAX]);
  // cond() evaluated at loop top: stop => discard speculative step (no flip)
  if (t > MAX_TIME || relax < RELAX_TOL) { ctrl[C_DONE] = 1u; return; }

  const float dt  = __uint_as_float(ctrl[C_DT]);
  const float err = __uint_as_float(ctrl[C_ERR]);
  const bool accept = (err < RK_TOL) || (dt <= MIN_DT);
  if (accept) {
    ctrl[C_T]   = __float_as_uint(t + dt);
    ctrl[C_CUR] ^= 1u;                 // accepted x5 becomes current state
  }
  float dtn = 0.9f * dt * powf(RK_TOL / (err + 1e-12f), 0.2f);
  dtn = fminf(fmaxf(dtn, MIN_DT), MAX_DT);
  ctrl[C_DT]    = __float_as_uint(dtn);
  ctrl[C_ERR]   = 0u;
  ctrl[C_RELAX] = 0u;
}

// ---------------------------------------------------------------- finalize

__global__ void finalize(const unsigned* __restrict__ ctrl,
                         const float4* __restrict__ buf0,
                         float4* __restrict__ out, long n4) {
  if (ctrl[C_CUR] != 0u) return;       // result already lives in d_out (buf1)
  long i = (long)blockIdx.x * blockDim.x + threadIdx.x;
  if (i < n4) out[i] = buf0[i];
}

// ---------------------------------------------------------------- launch

extern "C" void kernel_launch(void* const* d_in, const int* in_sizes, int n_in,
                              void* d_out, int out_size, void* d_ws, size_t ws_size,
                              hipStream_t stream) {
  (void)in_sizes; (void)n_in; (void)out_size; (void)ws_size;
  const float* x = (const float*)d_in[0];
  const float* W = (const float*)d_in[1];
  const float* b = (const float*)d_in[2];
  float* out = (float*)d_out;

  char* ws = (char*)d_ws;
  unsigned* ctrl = (unsigned*)ws;                                   // 256 B
  float2* P   = (float2*)(ws + 256);                                // 256 KB packed Ws
  float* buf0 = (float*)(ws + 256 + (size_t)UNITS * UNITS * sizeof(float));
  // buf1 == d_out (ping-pong partner)

  const long n4 = (long)BATCH * UNITS / 4;
  const int  cpblocks = (int)(n4 / 256);

  init_ctrl<<<1, 1, 0, stream>>>(ctrl);
  prep_pack<<<(UNITS * UNITS / 2) / 256, 256, 0, stream>>>(W, P);
  copy_in<<<cpblocks, 256, 0, stream>>>((const float4*)x, (float4*)buf0, n4);

  for (int s = 0; s < MAX_STEPS; ++s) {
    hopfield_rkf_step<<<BATCH / MTILE, 256, 0, stream>>>(P, b, buf0, out, ctrl);
    rkf_control<<<1, 1, 0, stream>>>(ctrl);
  }

  finalize<<<cpblocks, 256, 0, stream>>>(ctrl, (const float4*)buf0, (float4*)out, n4);
}